// LaplaceProductKernel_21182778703979
// MI455X (gfx1250) — compile-verified
//
#include <hip/hip_runtime.h>
#include <stdint.h>

// ---------------------------------------------------------------------------
// Laplace kernel  out[i,j] = exp(-max(sum_k |x1[i,k]-x2[j,k]| / 64, 1e-15))
// (mean-adjustment cancels in the difference; lengthscale == d == 64)
//
// MI455X: VALU-bound (v_sub_f32 + v_add_f32 |abs|, VOPD dual-issue capable).
// Tiles staged by the CDNA5 Tensor Data Mover with 4-dword pad insertion
// (row pitch 68 dwords -> 16B-aligned rows -> ds_load_b128 across k).
// Block = 256 threads, output tile 128x64, micro-tile 8x4 per thread.
// ---------------------------------------------------------------------------

typedef unsigned int v4u __attribute__((ext_vector_type(4)));
typedef int          v4i __attribute__((ext_vector_type(4)));
typedef int          v8i __attribute__((ext_vector_type(8)));

#define TM        128  // x1 rows per block tile
#define TN        64   // x2 rows per block tile
#define DDIM      64
#define LDSPITCH  68   // 64 data dwords + 4 pad dwords (16B-aligned rows)

__device__ __forceinline__ unsigned lds_off(const void* p) {
  // flat shared pointer: low 32 bits are the workgroup-relative LDS offset
  return (unsigned)(uintptr_t)p;
}

#if __has_builtin(__builtin_amdgcn_tensor_load_to_lds)
#define HAVE_TDM 1
#else
#define HAVE_TDM 0
#endif

#if HAVE_TDM
// TDM load of a contiguous (rows x 64) f32 tile (row stride 64 elems) into
// LDS, inserting 4 pad dwords after every 64 data dwords -> pitch 68 dwords.
__device__ __forceinline__ void tdm_load_tile(unsigned lds_byte_off,
                                              unsigned long long gaddr,
                                              unsigned rows) {
  v4u g0;
  g0[0] = 1u;                                     // count=1, user descriptor
  g0[1] = lds_byte_off;                           // D#.lds_addr
  g0[2] = (unsigned)(gaddr & 0xFFFFFFFFull);      // global_addr[31:0]
  g0[3] = (unsigned)(gaddr >> 32) | (2u << 30);   // global_addr[56:32] | type=2

  v8i g1;
  // data_size=2 (4B), pad_enable=1, pad_interval=5 (64 dwords), pad_amount=3 (4 dwords)
  g1[0] = (int)((2u << 16) | (1u << 20) | (5u << 22) | (3u << 25));
  g1[1] = (int)(64u << 16);                       // tensor_dim0 = 64 (lo16)
  g1[2] = (int)((rows & 0xFFFFu) << 16);          // tensor_dim1 lo16
  g1[3] = (int)(64u << 16);                       // tile_dim0 = 64
  g1[4] = (int)rows;                              // tile_dim1 = rows
  g1[5] = 64;                                     // tensor_dim0_stride = 64 elems
  g1[6] = 0;
  g1[7] = 0;

  v4i z4 = {0, 0, 0, 0};
#if defined(__clang_major__) && (__clang_major__ >= 23)
  v8i z8 = {0, 0, 0, 0, 0, 0, 0, 0};
  __builtin_amdgcn_tensor_load_to_lds(g0, g1, z4, z4, z8, 0);
#else
  __builtin_amdgcn_tensor_load_to_lds(g0, g1, z4, z4, 0);
#endif
}
#endif

__global__ __launch_bounds__(256) void
laplace_l1_exp_kernel(const float* __restrict__ x1, const float* __restrict__ x2,
                      float* __restrict__ out, int n, int m) {
  __shared__ __align__(16) float s1[TM * LDSPITCH];  // 34816 B
  __shared__ __align__(16) float s2[TN * LDSPITCH];  // 17408 B

  const int t  = threadIdx.x;
  const int bc = blockIdx.x;   // column tile (x2 rows), width TN
  const int br = blockIdx.y;   // row tile    (x1 rows), height TM

#if HAVE_TDM
  if (t < 32) {  // wave 0 drives both DMA descriptors
    unsigned long long ga1 =
        (unsigned long long)(uintptr_t)(x1 + (size_t)br * TM * DDIM);
    unsigned long long ga2 =
        (unsigned long long)(uintptr_t)(x2 + (size_t)bc * TN * DDIM);
    tdm_load_tile(lds_off(s1), ga1, TM);
    tdm_load_tile(lds_off(s2), ga2, TN);
    __builtin_amdgcn_s_wait_tensorcnt(0);
  }
#else
  // Fallback staging: float4 global loads, padded LDS stores (pitch 68).
  for (int idx = t; idx < (TM * DDIM) / 4; idx += 256) {
    int row = idx >> 4;
    int k0  = (idx & 15) * 4;
    float4 v = *(const float4*)(x1 + ((size_t)br * TM + row) * DDIM + k0);
    *(float4*)&s1[row * LDSPITCH + k0] = v;
  }
  for (int idx = t; idx < (TN * DDIM) / 4; idx += 256) {
    int row = idx >> 4;
    int k0  = (idx & 15) * 4;
    float4 v = *(const float4*)(x2 + ((size_t)bc * TN + row) * DDIM + k0);
    *(float4*)&s2[row * LDSPITCH + k0] = v;
  }
#endif
  __syncthreads();

  // 16x16 thread grid, each thread owns an 8x4 micro-tile.
  const int tx = t & 15;       // column group: cols 4*tx .. 4*tx+3
  const int ty = t >> 4;       // row group:    rows 8*ty .. 8*ty+7
  const int r0 = ty * 8;
  const int c0 = tx * 4;

  const float* __restrict__ pa = &s1[r0 * LDSPITCH];
  const float* __restrict__ pb = &s2[c0 * LDSPITCH];

  float acc[8][4];
#pragma unroll
  for (int i = 0; i < 8; ++i)
#pragma unroll
    for (int j = 0; j < 4; ++j) acc[i][j] = 0.0f;

#pragma unroll
  for (int k0 = 0; k0 < DDIM; k0 += 4) {
    float4 bv[4];
#pragma unroll
    for (int j = 0; j < 4; ++j)
      bv[j] = *(const float4*)(pb + j * LDSPITCH + k0);   // ds_load_b128

#pragma unroll
    for (int i = 0; i < 8; ++i) {
      float4 av = *(const float4*)(pa + i * LDSPITCH + k0); // ds_load_b128
#pragma unroll
      for (int j = 0; j < 4; ++j) {
        acc[i][j] += fabsf(av.x - bv[j].x);
        acc[i][j] += fabsf(av.y - bv[j].y);
        acc[i][j] += fabsf(av.z - bv[j].z);
        acc[i][j] += fabsf(av.w - bv[j].w);
      }
    }
  }

  const float sc = 1.0f / 64.0f;  // 1/lengthscale (constant per dim)
  const int row0 = br * TM + r0;
  const int col0 = bc * TN + c0;
#pragma unroll
  for (int i = 0; i < 8; ++i) {
    float4 o;
    o.x = __expf(-fmaxf(acc[i][0] * sc, 1e-15f));
    o.y = __expf(-fmaxf(acc[i][1] * sc, 1e-15f));
    o.z = __expf(-fmaxf(acc[i][2] * sc, 1e-15f));
    o.w = __expf(-fmaxf(acc[i][3] * sc, 1e-15f));
    *(float4*)(out + (size_t)(row0 + i) * (size_t)m + col0) = o;
  }
}

extern "C" void kernel_launch(void* const* d_in, const int* in_sizes, int n_in,
                              void* d_out, int out_size, void* d_ws, size_t ws_size,
                              hipStream_t stream) {
  (void)n_in; (void)out_size; (void)d_ws; (void)ws_size;
  const float* x1 = (const float*)d_in[0];
  const float* x2 = (const float*)d_in[1];
  float* out = (float*)d_out;

  const int d = DDIM;
  const int n = in_sizes[0] / d;  // 8192
  const int m = in_sizes[1] / d;  // 8192

  dim3 grid(m / TN, n / TM);      // 128 x 64 tiles
  dim3 block(256);
  laplace_l1_exp_kernel<<<grid, block, 0, stream>>>(x1, x2, out, n, m);
}